// GraphEncoder_67534065762853
// MI455X (gfx1250) — compile-verified
//
#include <hip/hip_runtime.h>
#include <hip/hip_bf16.h>
#include <math.h>

// ---------------------------------------------------------------------------
// GraphEncoder on MI455X (gfx1250), fp32 WMMA path, branch-free batch padding.
//   B=8 graphs (padded to 16 WMMA columns), N=128 nodes, F=16 feats, T=4 steps.
// One workgroup per node-row i (recurrence is row-separable). Per-edge 16x16xF
// matmuls are 4 chained V_WMMA_F32_16X16X4_F32 (full fp32 precision).
// Activations use raw v_rcp_f32 (__builtin_amdgcn_rcpf) instead of IEEE div.
// ---------------------------------------------------------------------------

typedef __attribute__((ext_vector_type(2))) float v2f;
typedef __attribute__((ext_vector_type(4))) float v4f;
typedef __attribute__((ext_vector_type(8))) float v8f;

#define GN 128   // nodes
#define GF 16    // features
#define GB 8     // real batch
#define GP 16    // padded batch (WMMA columns)
#define GOUT 256
#define TSTEPS 4
#define NWAVES 8
#define NTHREADS 256

__device__ __forceinline__ float sigm(float x) {
  return __builtin_amdgcn_rcpf(1.0f + __expf(-x));       // v_exp + v_rcp
}
__device__ __forceinline__ float tanh_fast(float x) {
  float ex = __expf(2.0f * x);
  return 1.0f - 2.0f * __builtin_amdgcn_rcpf(ex + 1.0f); // (e^2x-1)/(e^2x+1)
}

// Chained 16x16x16 fp32 matmul-accumulate: D = A(16x16) * B(16x16) + C
__device__ __forceinline__ v8f wmma_acc(const v2f a[4], const v2f b[4], v8f c) {
#pragma unroll
  for (int m = 0; m < 4; ++m)
    c = __builtin_amdgcn_wmma_f32_16x16x4_f32(false, a[m], false, b[m],
                                              (short)0, c, false, false);
  return c;
}

// A fragment (row-major 16x16, M=row, K=col). ISA 16x4 layout:
//   VGPR0: K=4m+0 (lanes 0-15) / K=4m+2 (lanes 16-31); VGPR1: K=4m+1 / 4m+3.
__device__ __forceinline__ void load_a16(const float* __restrict__ A,
                                         int row, int half, v2f a[4]) {
#pragma unroll
  for (int m = 0; m < 4; ++m) {
    int k = 4 * m + 2 * half;
    a[m] = *(const v2f*)(A + row * GF + k);             // global_load_b64
  }
}

__global__ __launch_bounds__(NTHREADS)
void graph_enc_kernel(const float* __restrict__ x,    // [B,N,F]
                      const int*   __restrict__ adj,  // [B,N,N]
                      const float* __restrict__ Wu,   // [N,N,F,F]
                      const float* __restrict__ Wc,
                      const float* __restrict__ Uu,
                      const float* __restrict__ Uc,
                      const float* __restrict__ bu,   // [F]
                      const float* __restrict__ bc,   // [F]
                      const float* __restrict__ Un,   // [N,F,F]
                      const float* __restrict__ Um,   // [N,F,F]
                      float* __restrict__ enc)        // [B,N,F]
{
  __shared__ float s_msgs[GP][GN][GF];          // 128 KB, batches 8..15 = zero pad
  __shared__ float s_mrow[GP][GN];              // 8 KB, padded rows zero
  __shared__ float s_total[GP][GF];
  __shared__ float s_rmsum[GP][GF];
  __shared__ float s_xt[GP][GF];                // x tile [col=b][g], pad cols zero
  __shared__ float s_stA[NWAVES][GP][GF];       // staging [col][f] (b128 stores)
  __shared__ float s_stB[NWAVES][GP][GF];
  __shared__ float s_bu[GF], s_bc[GF];
  __shared__ float s_deg[GP];

  const int i    = blockIdx.x;       // node row
  const int tid  = threadIdx.x;
  const int wave = tid >> 5;
  const int lane = tid & 31;
  const int lcol = lane & 15;        // WMMA column = (padded) batch index
  const int half = lane >> 4;        // row-half selector
  const int fbase = 8 * half;

  // ---- init LDS (padding rows zeroed) ----
  for (int t = tid; t < GP * GN * GF; t += NTHREADS) (&s_msgs[0][0][0])[t] = 0.0f;
  for (int t = tid; t < GP * GN; t += NTHREADS) {
    int b = t / GN, j = t % GN;
    s_mrow[b][j] =
        (b < GB && adj[((size_t)b * GN + i) * GN + j] != 0) ? 1.0f : 0.0f;
  }
  if (tid < GF) { s_bu[tid] = bu[tid]; s_bc[tid] = bc[tid]; }
  {
    int col = tid >> 4, g = tid & 15;   // covers all 256 = 16x16
    s_xt[col][g] = (col < GB) ? x[((size_t)col * GN + i) * GF + g] : 0.0f;
  }
  __syncthreads();
  if (tid < GP) {
    float d = 0.0f;
    for (int j = 0; j < GN; ++j) d += s_mrow[tid][j];
    s_deg[tid] = d;                     // padding rows -> 0
  }
  __syncthreads();

  for (int t = 0; t < TSTEPS; ++t) {
    // ---- total[b][f] = sum_j mask*msgs ; reset rmsum (all 256 threads) ----
    {
      int b = tid >> 4, f = tid & 15;
      float s = 0.0f;
      for (int j = 0; j < GN; ++j) s += s_mrow[b][j] * s_msgs[b][j][f];
      s_total[b][f] = s;
      s_rmsum[b][f] = 0.0f;
    }
    __syncthreads();

    // ---- PASS 1: r gate, accumulate rmsum = sum_j mask*r*msgs ----
    for (int e = 0; e < GN / NWAVES; ++e) {
      int j = e * NWAVES + wave;
      const float* pWu = Wu + ((size_t)(i * GN + j)) * GF * GF;
      const float* pUu = Uu + ((size_t)(i * GN + j)) * GF * GF;
      if (e + 1 < GN / NWAVES) {   // prefetch next edge's tiles
        __builtin_prefetch(pUu + NWAVES * GF * GF, 0, 1);
        __builtin_prefetch(pWu + NWAVES * GF * GF, 0, 1);
      }
      v2f aU[4], aW[4], bX[4], bM[4];
      load_a16(pUu, lcol, half, aU);
      load_a16(pWu, lcol, half, aW);
#pragma unroll
      for (int m = 0; m < 4; ++m) {
        int k = 4 * m + 2 * half;
        bX[m] = *(const v2f*)(&s_xt[lcol][k]);          // ds_load_b64
        bM[m] = *(const v2f*)(&s_msgs[lcol][j][k]);
      }
      v8f cWu = {};
      cWu = wmma_acc(aW, bX, cWu);          // Wu[i,j] @ x_i   (all columns)
      v8f dr = wmma_acc(aU, bM, cWu);       // + Uu[i,j] @ msgs_ij
      float mv = s_mrow[lcol][j];
#pragma unroll
      for (int r = 0; r < 8; ++r) {
        int f = r + fbase;
        float rr = truncf(sigm(dr[r] + s_bu[f]));       // .long() cast semantics
        float rm = mv * rr * s_msgs[lcol][j][f];
        atomicAdd(&s_rmsum[lcol][f], rm);               // padding adds 0.0
      }
    }
    __syncthreads();

    // ---- PASS 2: z gate, candidate memory (Uc applied twice), msgs update ----
    for (int e = 0; e < GN / NWAVES; ++e) {
      int j = e * NWAVES + wave;
      const float* pWu = Wu + ((size_t)(i * GN + j)) * GF * GF;
      const float* pWc = Wc + ((size_t)(i * GN + j)) * GF * GF;
      const float* pUu = Uu + ((size_t)(i * GN + j)) * GF * GF;
      const float* pUc = Uc + ((size_t)(i * GN + j)) * GF * GF;
      v2f aU[4], aW[4], aC[4], aUC[4], bX[4], bM[4], bMO[4], bS[4];
      load_a16(pUu, lcol, half, aU);
      load_a16(pWu, lcol, half, aW);
      load_a16(pWc, lcol, half, aC);
      load_a16(pUc, lcol, half, aUC);
      float mv  = s_mrow[lcol][j];
      float dgt = (s_deg[lcol] > 1.0f) ? 1.0f : 0.0f;   // branch-free guard
#pragma unroll
      for (int m = 0; m < 4; ++m) {
        int k = 4 * m + 2 * half;
        bX[m] = *(const v2f*)(&s_xt[lcol][k]);
        bM[m] = *(const v2f*)(&s_msgs[lcol][j][k]);
        v2f tt = *(const v2f*)(&s_total[lcol][k]);
        bMO[m].x = dgt * (tt.x - mv * bM[m].x);
        bMO[m].y = dgt * (tt.y - mv * bM[m].y);
      }
      v8f cWu = {}, cWc = {};
      cWu = wmma_acc(aW, bX, cWu);          // Wu @ x
      cWc = wmma_acc(aC, bX, cWc);          // Wc @ x
      v8f dz = wmma_acc(aU, bMO, cWu);      // z pre-activation
      v8f dr = wmma_acc(aU, bM, cWu);       // r pre-activation (recomputed)

      // msgs_reset (D layout) -> staging A, [col][f] so per-lane b128 stores
      {
        v4f lo, hi;  // f = fbase..fbase+7 contiguous
#pragma unroll
        for (int r = 0; r < 8; ++r) {
          int f = r + fbase;
          float rr = truncf(sigm(dr[r] + s_bu[f]));
          float rm = mv * rr * s_msgs[lcol][j][f];
          float v = s_rmsum[lcol][f] - rm;              // padding: 0 - 0
          if (r < 4) lo[r] = v; else hi[r - 4] = v;
        }
        *(v4f*)(&s_stA[wave][lcol][fbase])     = lo;    // fbase is 0 or 8
        *(v4f*)(&s_stA[wave][lcol][fbase + 4]) = hi;
      }
      __syncthreads();
#pragma unroll
      for (int m = 0; m < 4; ++m) {
        int k = 4 * m + 2 * half;
        bS[m] = *(const v2f*)(&s_stA[wave][lcol][k]);
      }
      v8f zero = {};
      v8f dpre = wmma_acc(aUC, bS, zero);   // pre = Uc @ msgs_reset
      {
        v4f lo, hi;
#pragma unroll
        for (int r = 0; r < 8; ++r) { if (r < 4) lo[r] = dpre[r]; else hi[r-4] = dpre[r]; }
        *(v4f*)(&s_stB[wave][lcol][fbase])     = lo;
        *(v4f*)(&s_stB[wave][lcol][fbase + 4]) = hi;
      }
      __syncthreads();
#pragma unroll
      for (int m = 0; m < 4; ++m) {
        int k = 4 * m + 2 * half;
        bS[m] = *(const v2f*)(&s_stB[wave][lcol][k]);
      }
      v8f dcm = wmma_acc(aUC, bS, cWc);     // cm pre-act = Uc @ pre + Wc@x

#pragma unroll
      for (int r = 0; r < 8; ++r) {
        int f = r + fbase;
        float zz = sigm(dz[r] + s_bu[f]);
        float cm = tanh_fast(dcm[r] + s_bc[f]);
        float mo = dgt * (s_total[lcol][f] - mv * s_msgs[lcol][j][f]);
        s_msgs[lcol][j][f] = mv * (zz * mo + (1.0f - zz) * cm);  // pad: mv=0
      }
      __syncthreads();
    }
    __syncthreads();
  }

  // ---- readout: enc[b,i,:] = relu(Un[i]@x + Um[i]@msg_sum) ----
  {
    int b = tid >> 4, f = tid & 15;
    float ms = 0.0f;
    for (int j = 0; j < GN; ++j) ms += s_msgs[b][j][f];
    s_total[b][f] = ms;     // reuse as msg_sum (padding rows harmless)
  }
  __syncthreads();
  if (tid < GB * GF) {
    int b = tid >> 4, f = tid & 15;
    const float* pUn = Un + (size_t)i * GF * GF + f * GF;
    const float* pUm = Um + (size_t)i * GF * GF + f * GF;
    float acc = 0.0f;
    for (int g = 0; g < GF; ++g)
      acc += pUn[g] * x[((size_t)b * GN + i) * GF + g] + pUm[g] * s_total[b][g];
    enc[((size_t)b * GN + i) * GF + f] = fmaxf(acc, 0.0f);
  }
}

// ---- final linear: out[b,o] = sigmoid(enc[b,:] . lin_w[o,:] + lin_b[o]) ----
__global__ __launch_bounds__(NTHREADS)
void linear_kernel(const float* __restrict__ enc,   // [B, N*F]
                   const float* __restrict__ lw,    // [OUT, N*F]
                   const float* __restrict__ lb,    // [OUT]
                   float* __restrict__ out)         // [B, OUT]
{
  int t = blockIdx.x * NTHREADS + threadIdx.x;      // 0..2047
  int b = t >> 8;
  int o = t & (GOUT - 1);
  const float4* e = (const float4*)(enc + (size_t)b * GN * GF);
  const float4* w = (const float4*)(lw + (size_t)o * GN * GF);
  float acc = lb[o];
#pragma unroll 4
  for (int k = 0; k < (GN * GF) / 4; ++k) {
    float4 ev = e[k], wv = w[k];
    acc += ev.x * wv.x + ev.y * wv.y + ev.z * wv.z + ev.w * wv.w;
  }
  out[t] = __builtin_amdgcn_rcpf(1.0f + __expf(-acc));
}

extern "C" void kernel_launch(void* const* d_in, const int* in_sizes, int n_in,
                              void* d_out, int out_size, void* d_ws, size_t ws_size,
                              hipStream_t stream) {
  const float* x   = (const float*)d_in[0];
  const int*   adj = (const int*)d_in[1];
  // d_in[2] = batch_size (== 8, static)
  const float* Wu  = (const float*)d_in[3];
  const float* Wc  = (const float*)d_in[4];
  const float* Uu  = (const float*)d_in[5];
  const float* Uc  = (const float*)d_in[6];
  const float* bu  = (const float*)d_in[7];
  const float* bc  = (const float*)d_in[8];
  const float* Un  = (const float*)d_in[9];
  const float* Um  = (const float*)d_in[10];
  const float* lw  = (const float*)d_in[11];
  const float* lb  = (const float*)d_in[12];

  float* enc = (float*)d_ws;   // [B,N,F] = 64 KB scratch

  graph_enc_kernel<<<GN, NTHREADS, 0, stream>>>(x, adj, Wu, Wc, Uu, Uc,
                                                bu, bc, Un, Um, enc);
  linear_kernel<<<(GB * GOUT) / NTHREADS, NTHREADS, 0, stream>>>(enc, lw, lb,
                                                                 (float*)d_out);
}